// CycleOverlapLoss_46033459478654
// MI455X (gfx1250) — compile-verified
//
#include <hip/hip_runtime.h>
#include <hip/hip_bf16.h>
#include <stdint.h>

#define BATCH 8
#define HH 600
#define WW 800
#define NPIX (HH * WW)
#define TPB 256
#define PPT 8
#define TILE (TPB * PPT)  // 2048 pixels per block
#define NWAVES (TPB / 32)

typedef unsigned int uint4v __attribute__((ext_vector_type(4)));
typedef int int8v  __attribute__((ext_vector_type(8)));
typedef int int4v  __attribute__((ext_vector_type(4)));

// ---- ordered-uint encoding so float min/max map to uint atomicMin/Max ----
__device__ __forceinline__ unsigned encf(float f) {
  unsigned u = __float_as_uint(f);
  return (u >> 31) ? ~u : (u | 0x80000000u);
}
__device__ __forceinline__ float decf(unsigned u) {
  return (u >> 31) ? __uint_as_float(u & 0x7FFFFFFFu) : __uint_as_float(~u);
}
// init values: enc(+inf)=0xFF800000 (min slots), enc(-inf)=0x007FFFFF (max slots)

__device__ void inv4x4(const float m[16], float out[16]) {
  float inv[16];
  inv[0]  =  m[5]*m[10]*m[15] - m[5]*m[11]*m[14] - m[9]*m[6]*m[15] + m[9]*m[7]*m[14] + m[13]*m[6]*m[11] - m[13]*m[7]*m[10];
  inv[4]  = -m[4]*m[10]*m[15] + m[4]*m[11]*m[14] + m[8]*m[6]*m[15] - m[8]*m[7]*m[14] - m[12]*m[6]*m[11] + m[12]*m[7]*m[10];
  inv[8]  =  m[4]*m[9]*m[15]  - m[4]*m[11]*m[13] - m[8]*m[5]*m[15] + m[8]*m[7]*m[13] + m[12]*m[5]*m[11] - m[12]*m[7]*m[9];
  inv[12] = -m[4]*m[9]*m[14]  + m[4]*m[10]*m[13] + m[8]*m[5]*m[14] - m[8]*m[6]*m[13] - m[12]*m[5]*m[10] + m[12]*m[6]*m[9];
  inv[1]  = -m[1]*m[10]*m[15] + m[1]*m[11]*m[14] + m[9]*m[2]*m[15] - m[9]*m[3]*m[14] - m[13]*m[2]*m[11] + m[13]*m[3]*m[10];
  inv[5]  =  m[0]*m[10]*m[15] - m[0]*m[11]*m[14] - m[8]*m[2]*m[15] + m[8]*m[3]*m[14] + m[12]*m[2]*m[11] - m[12]*m[3]*m[10];
  inv[9]  = -m[0]*m[9]*m[15]  + m[0]*m[11]*m[13] + m[8]*m[1]*m[15] - m[8]*m[3]*m[13] - m[12]*m[1]*m[11] + m[12]*m[3]*m[9];
  inv[13] =  m[0]*m[9]*m[14]  - m[0]*m[10]*m[13] - m[8]*m[1]*m[14] + m[8]*m[2]*m[13] + m[12]*m[1]*m[10] - m[12]*m[2]*m[9];
  inv[2]  =  m[1]*m[6]*m[15]  - m[1]*m[7]*m[14]  - m[5]*m[2]*m[15] + m[5]*m[3]*m[14] + m[13]*m[2]*m[7]  - m[13]*m[3]*m[6];
  inv[6]  = -m[0]*m[6]*m[15]  + m[0]*m[7]*m[14]  + m[4]*m[2]*m[15] - m[4]*m[3]*m[14] - m[12]*m[2]*m[7]  + m[12]*m[3]*m[6];
  inv[10] =  m[0]*m[5]*m[15]  - m[0]*m[7]*m[13]  - m[4]*m[1]*m[15] + m[4]*m[3]*m[13] + m[12]*m[1]*m[7]  - m[12]*m[3]*m[5];
  inv[14] = -m[0]*m[5]*m[14]  + m[0]*m[6]*m[13]  + m[4]*m[1]*m[14] - m[4]*m[2]*m[13] - m[12]*m[1]*m[6]  + m[12]*m[2]*m[5];
  inv[3]  = -m[1]*m[6]*m[11]  + m[1]*m[7]*m[10]  + m[5]*m[2]*m[11] - m[5]*m[3]*m[10] - m[9]*m[2]*m[7]   + m[9]*m[3]*m[6];
  inv[7]  =  m[0]*m[6]*m[11]  - m[0]*m[7]*m[10]  - m[4]*m[2]*m[11] + m[4]*m[3]*m[10] + m[8]*m[2]*m[7]   - m[8]*m[3]*m[6];
  inv[11] = -m[0]*m[5]*m[11]  + m[0]*m[7]*m[9]   + m[4]*m[1]*m[11] - m[4]*m[3]*m[9]  - m[8]*m[1]*m[7]   + m[8]*m[3]*m[5];
  inv[15] =  m[0]*m[5]*m[10]  - m[0]*m[6]*m[9]   - m[4]*m[1]*m[10] + m[4]*m[2]*m[9]  + m[8]*m[1]*m[6]   - m[8]*m[2]*m[5];
  float det = m[0]*inv[0] + m[1]*inv[4] + m[2]*inv[8] + m[3]*inv[12];
  float id = 1.0f / det;
  for (int i = 0; i < 16; ++i) out[i] = inv[i] * id;
}

// Per (direction, batch): fold the whole unproject->transform->reproject chain
// into 20 coefficients + the source bbox. Also (re)initialize atomic slots.
__global__ void col_prep(const float* __restrict__ pred1, const float* __restrict__ K1,
                         const float* __restrict__ pose1, const float* __restrict__ bbox1,
                         const float* __restrict__ ratio1,
                         const float* __restrict__ pred2, const float* __restrict__ K2,
                         const float* __restrict__ pose2, const float* __restrict__ bbox2,
                         const float* __restrict__ ratio2,
                         float* __restrict__ coefs, unsigned* __restrict__ red) {
  int t = threadIdx.x;
  if (t < 16 * 4) {
    int j = t & 3;
    red[t] = (j < 2) ? 0xFF800000u : 0x007FFFFFu;  // min slots / max slots
  }
  if (t >= 16) return;
  int d = t >> 3, b = t & 7;
  const float *Ka, *pa, *ba, *ra, *pra, *Kb_, *pb, *bb, *rb;
  if (d == 0) {
    Ka = K1 + 9*b; pa = pose1 + 16*b; ba = bbox1 + 2*b; ra = ratio1 + 2*b; pra = pred1 + 4*b;
    Kb_ = K2 + 9*b; pb = pose2 + 16*b; bb = bbox2 + 2*b; rb = ratio2 + 2*b;
  } else {
    Ka = K2 + 9*b; pa = pose2 + 16*b; ba = bbox2 + 2*b; ra = ratio2 + 2*b; pra = pred2 + 4*b;
    Kb_ = K1 + 9*b; pb = pose1 + 16*b; bb = bbox1 + 2*b; rb = ratio1 + 2*b;
  }
  float ma[16], inv[16], T[16];
  for (int i = 0; i < 16; ++i) ma[i] = pa[i];
  inv4x4(ma, inv);
  for (int r = 0; r < 4; ++r)
    for (int c = 0; c < 4; ++c) {
      float s = 0.f;
      for (int k = 0; k < 4; ++k) s += pb[r*4 + k] * inv[k*4 + c];
      T[r*4 + c] = s;
    }
  float fxa = Ka[0], cxa = Ka[2], fya = Ka[4], cya = Ka[5];
  float rya = ra[0], rxa = ra[1], bya = ba[0], bxa = ba[1];
  float ax = 1.0f / (rxa * fxa);
  float bx = ((bxa + 0.5f) / rxa - cxa) / fxa;
  float ay = 1.0f / (rya * fya);
  float by = ((bya + 0.5f) / rya - cya) / fya;
  float* C = coefs + t * 32;
  for (int i = 0; i < 4; ++i) {
    C[i]      = T[i*4+0] * ax;                               // A_i (x coef)
    C[4 + i]  = T[i*4+1] * ay;                               // B_i (y coef)
    C[8 + i]  = T[i*4+0] * bx + T[i*4+1] * by + T[i*4+2];    // C_i (const * z)
    C[12 + i] = T[i*4+3];                                    // D_i (translation)
  }
  float fxb = Kb_[0], cxb = Kb_[2], fyb = Kb_[4], cyb = Kb_[5];
  float ryb = rb[0], rxb = rb[1], byb = bb[0], bxb = bb[1];
  C[16] = fxb * rxb;                  // Gu
  C[17] = cxb * rxb - bxb - 0.5f;     // Hu
  C[18] = fyb * ryb;                  // Gv
  C[19] = cyb * ryb - byb - 0.5f;     // Hv
  C[20] = pra[0]; C[21] = pra[1]; C[22] = pra[2]; C[23] = pra[3];
}

__global__ __launch_bounds__(TPB)
void col_warp_reduce(const float* __restrict__ depth1, const float* __restrict__ depth2,
                     const float* __restrict__ coefs, unsigned* __restrict__ red) {
  const int d = blockIdx.z, bb = blockIdx.y;
  const float* da = ((d == 0) ? depth1 : depth2) + bb * NPIX;
  const float* db = ((d == 0) ? depth2 : depth1) + bb * NPIX;
  const int combo = d * 8 + bb;
  const float* C = coefs + combo * 32;
  const int start = blockIdx.x * TILE;

  __shared__ float sdep[TILE];
  __shared__ float swred[4][NWAVES];

  // --- Stage this block's contiguous depth_a chunk into LDS via the TDM ---
  if (threadIdx.x < 32) {  // one wave issues the DMA
    unsigned ldsoff = (unsigned)(uintptr_t)&sdep[0];  // low 32 bits = LDS offset
    unsigned long long gaddr = (unsigned long long)(uintptr_t)(da + start);
    unsigned rem = (unsigned)(NPIX - start);  // OOB reads return 0 -> masked by depth>0
    uint4v g0;
    g0[0] = 1u;                                           // count=1, user descriptor
    g0[1] = ldsoff;                                       // lds_addr
    g0[2] = (unsigned)(gaddr & 0xFFFFFFFFull);            // global_addr[31:0]
    g0[3] = (unsigned)((gaddr >> 32) & 0x1FFFFFFull) | (2u << 30);  // addr[56:32] | type=2
    int8v g1;
    g1[0] = (int)(2u << 16);                              // data_size=4B, no mask/flags
    g1[1] = (int)((rem & 0xFFFFu) << 16);                 // tensor_dim0[15:0]
    g1[2] = (int)((rem >> 16) | (1u << 16));              // tensor_dim0[31:16], tensor_dim1=1
    g1[3] = (int)((unsigned)TILE << 16);                  // tile_dim0 = 2048
    g1[4] = 1;                                            // tile_dim1 = 1
    g1[5] = (int)TILE;                                    // tensor_dim0_stride (unused, 1 row)
    g1[6] = 0;
    g1[7] = 0;
    int4v gz4 = {0, 0, 0, 0};
    int8v gz8 = {0, 0, 0, 0, 0, 0, 0, 0};
    __builtin_amdgcn_tensor_load_to_lds(g0, g1, gz4, gz4, gz8, 0);
    __builtin_amdgcn_s_wait_tensorcnt(0);
  }
  __syncthreads();

  const float A0 = C[0],  A1 = C[1],  A2 = C[2],  A3 = C[3];
  const float B0 = C[4],  B1 = C[5],  B2 = C[6],  B3 = C[7];
  const float C0 = C[8],  C1 = C[9],  C2 = C[10], C3 = C[11];
  const float D0 = C[12], D1 = C[13], D2 = C[14], D3 = C[15];
  const float Gu = C[16], Hu = C[17], Gv = C[18], Hv = C[19];
  const float p0 = C[20], p1 = C[21], p2 = C[22], p3 = C[23];

  const float INF = __uint_as_float(0x7F800000u);
  float umin = INF, vmin = INF, umax = -INF, vmax = -INF;

  for (int k = 0; k < PPT; ++k) {
    int lp = k * TPB + threadIdx.x;
    int p = start + lp;
    float z = sdep[lp];
    if (!(z == z)) z = 0.f;  // nan_to_num
    int row = p / WW;
    int col = p - row * WW;
    float x = (float)col, y = (float)row;
    bool src = (z > 0.f) && (x > p0) && (y > p1) && (x < p2) && (y < p3) && (p < NPIX);

    float q0 = fmaf(z, fmaf(A0, x, fmaf(B0, y, C0)), D0);
    float q1 = fmaf(z, fmaf(A1, x, fmaf(B1, y, C1)), D1);
    float q2 = fmaf(z, fmaf(A2, x, fmaf(B2, y, C2)), D2);
    float q3 = fmaf(z, fmaf(A3, x, fmaf(B3, y, C3)), D3);
    float iq2 = 1.0f / q2;
    float u2 = fmaf(Gu * q0, iq2, Hu);
    float v2 = fmaf(Gv * q1, iq2, Hv);
    float z2 = q2 / q3;

    float fi0 = floorf(v2), fi1 = ceilf(v2);
    float fj0 = floorf(u2), fj1 = ceilf(u2);
    bool vc = (fi0 >= 0.f) && (fj0 >= 0.f) && (fi1 < (float)HH) && (fj1 < (float)WW);
    float fi0c = fminf(fmaxf(fi0, 0.f), (float)(HH - 1));
    float fi1c = fminf(fmaxf(fi1, 0.f), (float)(HH - 1));
    float fj0c = fminf(fmaxf(fj0, 0.f), (float)(WW - 1));
    float fj1c = fminf(fmaxf(fj1, 0.f), (float)(WW - 1));
    int i0 = (int)fi0c, i1 = (int)fi1c, j0 = (int)fj0c, j1 = (int)fj1c;
    float d00 = db[i0 * WW + j0], d01 = db[i0 * WW + j1];
    float d10 = db[i1 * WW + j0], d11 = db[i1 * WW + j1];
    bool vd = (d00 > 0.f) && (d01 > 0.f) && (d10 > 0.f) && (d11 > 0.f);
    float di = v2 - fi0, dj = u2 - fj0;
    float interp = (1.f - di) * (1.f - dj) * d00 + (1.f - di) * dj * d01 +
                   di * (1.f - dj) * d10 + di * dj * d11;
    bool inl = fabsf(z2 - interp) < 1.0f;
    bool full = src && vc && vd && inl;
    if (full) {
      umin = fminf(umin, u2); umax = fmaxf(umax, u2);
      vmin = fminf(vmin, v2); vmax = fmaxf(vmax, v2);
    }
  }

  // wave32 butterfly reduction
  for (int off = 16; off > 0; off >>= 1) {
    umin = fminf(umin, __shfl_xor(umin, off));
    vmin = fminf(vmin, __shfl_xor(vmin, off));
    umax = fmaxf(umax, __shfl_xor(umax, off));
    vmax = fmaxf(vmax, __shfl_xor(vmax, off));
  }
  // block-level reduction in LDS: only 4 atomics per block
  int wid = threadIdx.x >> 5;
  if ((threadIdx.x & 31) == 0) {
    swred[0][wid] = umin; swred[1][wid] = vmin;
    swred[2][wid] = umax; swred[3][wid] = vmax;
  }
  __syncthreads();
  if (threadIdx.x == 0) {
    float a = swred[0][0], bmin = swred[1][0], c = swred[2][0], e = swred[3][0];
    for (int i = 1; i < NWAVES; ++i) {
      a = fminf(a, swred[0][i]); bmin = fminf(bmin, swred[1][i]);
      c = fmaxf(c, swred[2][i]); e = fmaxf(e, swred[3][i]);
    }
    unsigned* s = red + combo * 4;
    atomicMin(s + 0, encf(a));
    atomicMin(s + 1, encf(bmin));
    atomicMax(s + 2, encf(c));
    atomicMax(s + 3, encf(e));
  }
}

__device__ float giou_loss(const float* p, const float* t) {
  float lt0 = fmaxf(p[0], t[0]), lt1 = fmaxf(p[1], t[1]);
  float rb0 = fminf(p[2], t[2]), rb1 = fminf(p[3], t[3]);
  float w = fmaxf(rb0 - lt0, 0.f), h = fmaxf(rb1 - lt1, 0.f);
  float ov = w * h;
  float ap = (p[2] - p[0]) * (p[3] - p[1]);
  float ag = (t[2] - t[0]) * (t[3] - t[1]);
  float un = ap + ag - ov + 1e-7f;
  float iou = ov / un;
  float e10 = fminf(p[0], t[0]), e11 = fminf(p[1], t[1]);
  float e20 = fmaxf(p[2], t[2]), e21 = fmaxf(p[3], t[3]);
  float ew = fmaxf(e20 - e10, 0.f), eh = fmaxf(e21 - e11, 0.f);
  float ea = ew * eh + 1e-7f;
  float g = iou - (ea - un) / ea;
  return 1.f - g;
}

__global__ void col_finalize(const float* __restrict__ pred1, const float* __restrict__ pred2,
                             const unsigned* __restrict__ red, float* __restrict__ out) {
  int b = threadIdx.x;
  if (b >= BATCH) return;
  float w2box[4], w1box[4];
  {  // combo d=0: points of image1 warped into image2 -> compared against pred2
    const unsigned* s = red + (0 * 8 + b) * 4;
    w2box[0] = fminf(fmaxf(decf(s[0]), 0.f), (float)WW);
    w2box[1] = fminf(fmaxf(decf(s[1]), 0.f), (float)HH);
    w2box[2] = fminf(fmaxf(decf(s[2]), 0.f), (float)WW);
    w2box[3] = fminf(fmaxf(decf(s[3]), 0.f), (float)HH);
  }
  {  // combo d=1: points of image2 warped into image1 -> compared against pred1
    const unsigned* s = red + (1 * 8 + b) * 4;
    w1box[0] = fminf(fmaxf(decf(s[0]), 0.f), (float)WW);
    w1box[1] = fminf(fmaxf(decf(s[1]), 0.f), (float)HH);
    w1box[2] = fminf(fmaxf(decf(s[2]), 0.f), (float)WW);
    w1box[3] = fminf(fmaxf(decf(s[3]), 0.f), (float)HH);
  }
  float loss1 = giou_loss(pred1 + 4 * b, w1box);
  float loss2 = giou_loss(pred2 + 4 * b, w2box);
  out[b] = 0.5f * (loss1 + loss2);
}

extern "C" void kernel_launch(void* const* d_in, const int* in_sizes, int n_in,
                              void* d_out, int out_size, void* d_ws, size_t ws_size,
                              hipStream_t stream) {
  const float* pred1  = (const float*)d_in[0];
  const float* depth1 = (const float*)d_in[1];
  const float* K1     = (const float*)d_in[2];
  const float* pose1  = (const float*)d_in[3];
  const float* bbox1  = (const float*)d_in[4];
  const float* ratio1 = (const float*)d_in[5];
  const float* pred2  = (const float*)d_in[6];
  const float* depth2 = (const float*)d_in[7];
  const float* K2     = (const float*)d_in[8];
  const float* pose2  = (const float*)d_in[9];
  const float* bbox2  = (const float*)d_in[10];
  const float* ratio2 = (const float*)d_in[11];
  float* out = (float*)d_out;

  float* coefs = (float*)d_ws;                          // 16 combos * 32 floats = 2048 B
  unsigned* red = (unsigned*)((char*)d_ws + 2048);      // 16 combos * 4 uints  = 256 B

  col_prep<<<1, 64, 0, stream>>>(pred1, K1, pose1, bbox1, ratio1,
                                 pred2, K2, pose2, bbox2, ratio2, coefs, red);
  dim3 grid((NPIX + TILE - 1) / TILE, BATCH, 2);
  col_warp_reduce<<<grid, TPB, 0, stream>>>(depth1, depth2, coefs, red);
  col_finalize<<<1, 32, 0, stream>>>(pred1, pred2, red, out);
}